// R1C1_78108275245417
// MI455X (gfx1250) — compile-verified
//
#include <hip/hip_runtime.h>

typedef float v2f __attribute__((ext_vector_type(2)));
typedef float v4f __attribute__((ext_vector_type(4)));
typedef float v8f __attribute__((ext_vector_type(8)));

constexpr int S_LEN = 4096;   // steps per row
constexpr int NB    = 8192;   // batch rows
constexpr int NC    = 16;     // chunks along S
constexpr int CL    = 256;    // steps per chunk (NC*CL == S_LEN)

#define DEV static __device__ __forceinline__

DEV float sigmoidf(float x) { return 1.f / (1.f + __expf(-x)); }
DEV float boundedf(float raw, float lo, float hi) { return lo + (hi - lo) * sigmoidf(raw); }

// integer power by squaring: robust for any sign of base, e <= 256
DEV float ipowf(float b, int e) {
    float r = 1.f, m = b;
    while (e > 0) { if (e & 1) r *= m; m *= m; e >>= 1; }
    return r;
}

struct Params { float a, kc, invR, g, Aeff; };
DEV Params get_params(const float* rR, const float* rC, const float* rA, const float* rG) {
    float R = boundedf(*rR, 1e-4f, 0.2f);
    float C = boundedf(*rC, 1e5f, 1e8f);
    float A = boundedf(*rA, 0.f, 0.2f);
    float g = boundedf(*rG, 1.f, 20000.f);
    Params p;
    p.invR = 1.f / R;
    p.kc   = 900.f / C;          // DELTA_T / C
    p.a    = 1.f - p.kc * p.invR;
    p.g    = g;
    p.Aeff = A;
    return p;
}

// -------------------------------------------------------------------------
// Kernel 1: per-(16-row group, chunk) WMMA scan with carry=0 for chunks>0.
// One wave32 owns 16 batch rows x 256 steps; each 16-step tile is
//   D(16x16) = L(16x16) * Bm(16x16) + a^{i+1}*carry,
// done as 4x V_WMMA_F32_16X16X4_F32 (K-chunks of 4).
// -------------------------------------------------------------------------
__global__ __launch_bounds__(256) void rc_scan_kernel(
    const float* __restrict__ in,
    const float* rR, const float* rC, const float* rA, const float* rG,
    float* __restrict__ out, float* __restrict__ lastbuf)
{
    const int warp  = threadIdx.x >> 5;
    const int lane  = threadIdx.x & 31;
    const int g     = blockIdx.x * 8 + warp;   // 8192 waves total
    const int bg    = g & 511;                 // batch group (16 rows)
    const int chunk = g >> 9;                  // 0..15
    const int col   = lane & 15;               // N (and M index for A operand)
    const int hi    = lane >> 4;

    const Params P = get_params(rR, rC, rA, rG);
    const float a = P.a;

    // A operand: L[M][j] = (j<=M) ? a^(M-j) : 0 ; 16x4 f32 layout:
    // lanes 0-15 hold K = {kb, kb+1}, lanes 16-31 hold K = {kb+2, kb+3}
    v2f Aop[4];
#pragma unroll
    for (int c4 = 0; c4 < 4; ++c4) {
        int kb = 4 * c4 + 2 * hi;
        Aop[c4].x = (col >= kb    ) ? ipowf(a, col - kb    ) : 0.f;
        Aop[c4].y = (col >= kb + 1) ? ipowf(a, col - kb - 1) : 0.f;
    }
    // carry scaling per accumulator VGPR r: a^{M+1}, M = r + 8*hi
    float cs[8];
#pragma unroll
    for (int r = 0; r < 8; ++r) cs[r] = ipowf(a, r + 8 * hi + 1);

    const int    row   = bg * 16 + col;
    const float* rowp  = in + (size_t)row * S_LEN * 4;
    float*       obase = out + (size_t)row * S_LEN;

    float carry = (chunk == 0) ? rowp[0] : 0.f;  // t0 = input[row][0][0]

    int s0 = chunk * CL;
#pragma unroll 1
    for (int t = 0; t < CL / 16; ++t) {
        const float* p = rowp + (size_t)s0 * 4;

        // B operand: Bm[K][N] = b at step K of batch column N; mirrors A layout.
        v2f Bop[4];
#pragma unroll
        for (int c4 = 0; c4 < 4; ++c4) {
            int kb = 4 * c4 + 2 * hi;
            v4f x0 = __builtin_nontemporal_load((const v4f*)(p + kb * 4));
            v4f x1 = __builtin_nontemporal_load((const v4f*)(p + kb * 4 + 4));
            // b = (DT/C) * (t_out/R - g*u + A_eff*solar)   (COOLING_SIGN = -1)
            Bop[c4].x = P.kc * (x0.y * P.invR - P.g * x0.z + P.Aeff * x0.w);
            Bop[c4].y = P.kc * (x1.y * P.invR - P.g * x1.z + P.Aeff * x1.w);
        }

        v8f acc;
#pragma unroll
        for (int r = 0; r < 8; ++r) acc[r] = cs[r] * carry;

#pragma unroll
        for (int c4 = 0; c4 < 4; ++c4)
            acc = __builtin_amdgcn_wmma_f32_16x16x4_f32(
                false, Aop[c4], false, Bop[c4], (short)0, acc, false, false);

        // new carry = D[M=15][N=col] : VGPR 7, lane 16+col
        carry = __shfl(acc[7], 16 + col, 32);

        // store: lane covers (N=col, M = 8*hi + r), contiguous along s
        v4f o0 = {acc[0], acc[1], acc[2], acc[3]};
        v4f o1 = {acc[4], acc[5], acc[6], acc[7]};
        *(v4f*)(obase + s0 + 8 * hi)     = o0;
        *(v4f*)(obase + s0 + 8 * hi + 4) = o1;

        s0 += 16;
    }

    if (lane < 16) lastbuf[(size_t)row * NC + chunk] = carry; // local chunk-final value
}

// -------------------------------------------------------------------------
// Kernel 2: serial carry propagation across chunks (tiny: 8192 threads x 15).
// true_carry(1) = last(0);  true_carry(j+1) = last(j) + a^CL * true_carry(j)
// -------------------------------------------------------------------------
__global__ void rc_carry_kernel(
    const float* rR, const float* rC, const float* rA, const float* rG,
    const float* __restrict__ lastbuf, float* __restrict__ carrybuf)
{
    int n = blockIdx.x * blockDim.x + threadIdx.x;
    if (n >= NB) return;
    const Params P = get_params(rR, rC, rA, rG);
    const float aL = ipowf(P.a, CL);
    const float* l  = lastbuf + (size_t)n * NC;
    float*       cb = carrybuf + (size_t)n * NC;
    float c = l[0];
    cb[1] = c;
#pragma unroll
    for (int j = 1; j < NC - 1; ++j) { c = l[j] + aL * c; cb[j + 1] = c; }
}

// -------------------------------------------------------------------------
// Kernel 3: fixup  out[n][s] += a^{(s mod CL)+1} * carry(chunk(s)),  s >= CL.
// Output is L2-resident (128 MB < 192 MB L2, k1 input loads were NT),
// so this extra pass is L2 traffic, not HBM. float4 per thread, coalesced.
// -------------------------------------------------------------------------
__global__ __launch_bounds__(256) void rc_fixup_kernel(
    const float* rR, const float* rC, const float* rA, const float* rG,
    const float* __restrict__ carrybuf, float* __restrict__ out)
{
    const unsigned t = blockIdx.x * 256u + threadIdx.x;   // NB * 960 threads
    const unsigned n = t / 960u;
    const unsigned q = t % 960u;
    if (n >= NB) return;
    const Params P = get_params(rR, rC, rA, rG);

    const int s  = CL + 4 * (int)q;          // [256, 4096), 16B aligned
    const int ci = s >> 8;                   // chunk index 1..15
    const float c = carrybuf[(size_t)n * NC + ci];

    float pw = ipowf(P.a, (s & (CL - 1)) + 1);
    v4f* op = (v4f*)(out + (size_t)n * S_LEN + s);
    v4f o = *op;
    o.x += pw * c; pw *= P.a;
    o.y += pw * c; pw *= P.a;
    o.z += pw * c; pw *= P.a;
    o.w += pw * c;
    *op = o;
}

extern "C" void kernel_launch(void* const* d_in, const int* in_sizes, int n_in,
                              void* d_out, int out_size, void* d_ws, size_t ws_size,
                              hipStream_t stream)
{
    const float* in = (const float*)d_in[0];
    const float* rR = (const float*)d_in[1];
    const float* rC = (const float*)d_in[2];
    const float* rA = (const float*)d_in[3];
    const float* rG = (const float*)d_in[4];
    float* out = (float*)d_out;

    float* lastbuf  = (float*)d_ws;                   // NB*NC floats
    float* carrybuf = lastbuf + (size_t)NB * NC;      // NB*NC floats

    // k1: 8192 waves, 8 waves/block
    rc_scan_kernel<<<1024, 256, 0, stream>>>(in, rR, rC, rA, rG, out, lastbuf);
    // k2: one thread per batch row
    rc_carry_kernel<<<(NB + 255) / 256, 256, 0, stream>>>(rR, rC, rA, rG, lastbuf, carrybuf);
    // k3: NB*960 float4-threads over chunks 1..15
    rc_fixup_kernel<<<(NB * 960) / 256, 256, 0, stream>>>(rR, rC, rA, rG, carrybuf, out);

    (void)in_sizes; (void)n_in; (void)out_size; (void)ws_size;
}